// ClassifierGridSearch_80178449482550
// MI455X (gfx1250) — compile-verified
//
#include <hip/hip_runtime.h>

typedef float v2f __attribute__((ext_vector_type(2)));
typedef float v8f __attribute__((ext_vector_type(8)));

#define RG_N_NODES 50000
#define RG_N_EDGES 800000
#define RG_N_REL 16
#define RG_N_GRAPHS 64
#define RG_IN_DIM 128
#define RG_HID 128
#define RG_FC 256
#define RG_N_CLASSES 16

// -------------------------------------------------------------------------
// zero fill
__global__ void k_zero(float* __restrict__ p, int n) {
    int i = blockIdx.x * blockDim.x + threadIdx.x;
    int stride = gridDim.x * blockDim.x;
    for (; i < n; i += stride) p[i] = 0.0f;
}

// -------------------------------------------------------------------------
// GEMM, K=128 fixed, N compile-time:  Cout = [relu]( (Cin?) + A @ B + (bias?) )
// A: [M,128] row-major. B: [128,N] row-major. One wave -> 16x64 C tile via
// V_WMMA_F32_16X16X4_F32 (exact fp32 path). Compile-time N makes every load an
// immediate offset from a loop-invariant base, and the explicit 1-step register
// pipeline (prefetch k+4 before issuing k's WMMAs) overlaps VMEM with WMMA.
// A-frag (ISA 7.12.2, 32-bit A 16x4): lanes 0-15 M=0..15 / VGPR0=K{0,2}, VGPR1=K{1,3}.
// B-frag mirrored (N across lanes). C/D: VGPR v holds M = v + 8*(lane>=16), N = lane%16.
template <int N>
__global__ void k_gemm_k128(const float* __restrict__ A, const float* __restrict__ B,
                            const float* __restrict__ Cin, float* __restrict__ Cout,
                            const float* __restrict__ bias, int M, int relu) {
    const int lane  = threadIdx.x & 31;
    const int wave  = threadIdx.x >> 5;
    const int w     = blockIdx.x * (blockDim.x >> 5) + wave;
    constexpr int tilesN = N >> 6;             // 64 columns per wave
    const int mTile = w / tilesN;
    const int nBase = (w % tilesN) << 6;
    if (mTile * 16 >= M) return;               // wave-uniform: EXEC stays all-1s

    const int lr   = lane & 15;
    const int half = lane >> 4;

    // loop-invariant bases; all k offsets are compile-time immediates
    const float* Ap = A + (size_t)(mTile * 16 + lr) * 128 + 2 * half;  // + k0
    const float* Bp = B + (size_t)(2 * half) * N + nBase + lr;         // + k0*N

    v8f acc[4] = { {}, {}, {}, {} };

    // stage 0 fragments
    v2f a0 = *(const v2f*)(Ap);
    v2f b0[4];
#pragma unroll
    for (int t = 0; t < 4; ++t) { b0[t].x = Bp[t * 16]; b0[t].y = Bp[N + t * 16]; }

#pragma unroll
    for (int k0 = 0; k0 < 128; k0 += 4) {
        v2f a1 = a0;
        v2f b1[4] = { b0[0], b0[1], b0[2], b0[3] };
        if (k0 + 4 < 128) {                    // compile-time under full unroll
            a1 = *(const v2f*)(Ap + (k0 + 4));
            const float* bp = Bp + (size_t)(k0 + 4) * N;
#pragma unroll
            for (int t = 0; t < 4; ++t) { b1[t].x = bp[t * 16]; b1[t].y = bp[N + t * 16]; }
        }
#pragma unroll
        for (int t = 0; t < 4; ++t) {
            acc[t] = __builtin_amdgcn_wmma_f32_16x16x4_f32(
                false, a0, false, b0[t], (short)0, acc[t], false, false);
        }
        a0 = a1;
#pragma unroll
        for (int t = 0; t < 4; ++t) b0[t] = b1[t];
    }

#pragma unroll
    for (int t = 0; t < 4; ++t) {
        const int n  = nBase + t * 16 + lr;
        const float bv = bias ? bias[n] : 0.0f;
#pragma unroll
        for (int v = 0; v < 8; ++v) {
            const int m = mTile * 16 + v + 8 * half;
            float x = acc[t][v] + bv;
            if (Cin)  x += Cin[(size_t)m * N + n];
            if (relu) x = fmaxf(x, 0.0f);
            Cout[(size_t)m * N + n] = x;
        }
    }
}

// -------------------------------------------------------------------------
// Edge scatter for one relation: agg[dst] += tmp[src] for edges with rel==r.
// Grid-stride over edges (fixed modest grid -> low launch overhead under capture).
// One wave per edge, float4 gather + 4 global f32 atomics per lane.
__global__ void k_scatter_rel(const float* __restrict__ tmp, const int* __restrict__ src,
                              const int* __restrict__ dst, const int* __restrict__ rel,
                              float* __restrict__ agg, int r, int nE) {
    const int lane = threadIdx.x & 31;
    const int w0   = (blockIdx.x * blockDim.x + threadIdx.x) >> 5;
    const int nW   = (gridDim.x * blockDim.x) >> 5;
    for (int e = w0; e < nE; e += nW) {
        if (rel[e] != r) continue;
        const int s = src[e];
        const int d = dst[e];
        const float4 x = ((const float4*)(tmp + (size_t)s * 128))[lane];
        float* pd = agg + (size_t)d * 128 + lane * 4;
        atomicAdd(pd + 0, x.x);
        atomicAdd(pd + 1, x.y);
        atomicAdd(pd + 2, x.z);
        atomicAdd(pd + 3, x.w);
    }
}

// -------------------------------------------------------------------------
// Mean pooling, pass 1: per-graph sums + counts. One wave per node, grid-stride.
__global__ void k_pool(const float* __restrict__ h, const int* __restrict__ gid,
                       float* __restrict__ sums, float* __restrict__ cnts, int nV) {
    const int lane = threadIdx.x & 31;
    const int w0   = (blockIdx.x * blockDim.x + threadIdx.x) >> 5;
    const int nW   = (gridDim.x * blockDim.x) >> 5;
    for (int v = w0; v < nV; v += nW) {
        const int g = gid[v];
        const float4 x = ((const float4*)(h + (size_t)v * 128))[lane];
        float* pd = sums + (size_t)g * 128 + lane * 4;
        atomicAdd(pd + 0, x.x);
        atomicAdd(pd + 1, x.y);
        atomicAdd(pd + 2, x.z);
        atomicAdd(pd + 3, x.w);
        if (lane == 0) atomicAdd(&cnts[g], 1.0f);
    }
}

// pass 2: divide
__global__ void k_pool_div(const float* __restrict__ sums, const float* __restrict__ cnts,
                           float* __restrict__ hg) {
    const int i = blockIdx.x * blockDim.x + threadIdx.x;
    if (i < RG_N_GRAPHS * RG_HID) hg[i] = sums[i] / fmaxf(cnts[i >> 7], 1.0f);
}

// -------------------------------------------------------------------------
// Classification head + softmax. One block (32 threads) per graph.
__global__ void k_head(const float* __restrict__ fc, const float* __restrict__ Wc,
                       const float* __restrict__ bc, float* __restrict__ out) {
    const int g = blockIdx.x;
    const int c = threadIdx.x;
    __shared__ float sl[RG_N_CLASSES];
    if (c < RG_N_CLASSES) {
        float acc = bc[c];
        const float* row = fc + (size_t)g * RG_FC;
        for (int k = 0; k < RG_FC; ++k) acc += row[k] * Wc[k * RG_N_CLASSES + c];
        sl[c] = acc;
    }
    __syncthreads();
    if (c < RG_N_CLASSES) {
        float m = sl[0];
        for (int i = 1; i < RG_N_CLASSES; ++i) m = fmaxf(m, sl[i]);
        float denom = 0.0f;
        for (int i = 0; i < RG_N_CLASSES; ++i) denom += expf(sl[i] - m);
        out[g * RG_N_CLASSES + c] = expf(sl[c] - m) / denom;
    }
}

// -------------------------------------------------------------------------
extern "C" void kernel_launch(void* const* d_in, const int* in_sizes, int n_in,
                              void* d_out, int out_size, void* d_ws, size_t ws_size,
                              hipStream_t stream) {
    const float* h   = (const float*)d_in[0];
    const int*   src = (const int*)d_in[1];
    const int*   dst = (const int*)d_in[2];
    const int*   rel = (const int*)d_in[3];
    const int*   gid = (const int*)d_in[4];
    const float* W1  = (const float*)d_in[5];
    const float* Ws1 = (const float*)d_in[6];
    const float* b1  = (const float*)d_in[7];
    const float* W2  = (const float*)d_in[8];
    const float* Ws2 = (const float*)d_in[9];
    const float* b2  = (const float*)d_in[10];
    const float* Wfc = (const float*)d_in[11];
    const float* bfc = (const float*)d_in[12];
    const float* Wc  = (const float*)d_in[13];
    const float* bc  = (const float*)d_in[14];
    float* out = (float*)d_out;

    // workspace layout (floats)
    float* ws = (float*)d_ws;
    const size_t NH = (size_t)RG_N_NODES * RG_HID;      // 6.4M floats
    float* tmp   = ws;                                  // per-relation transformed feats
    float* agg   = tmp + NH;                            // neighbor aggregation
    float* h1    = agg + NH;                            // layer-1 output
    float* sums  = h1 + NH;                             // [G,128]
    float* cnts  = sums + RG_N_GRAPHS * RG_HID;         // [G]
    float* hg    = cnts + RG_N_GRAPHS;                  // [G,128]
    float* fcout = hg + RG_N_GRAPHS * RG_HID;           // [G,256]

    const int gemmBlk = 256;  // 8 waves
    auto gemmGrid = [](int M, int N) {
        int waves = (M / 16) * (N / 64);
        return (waves + 7) / 8;
    };
    const int gridNodesGemm = gemmGrid(RG_N_NODES, RG_HID);
    const int gridScat      = 4096;   // fixed grid, grid-stride over 800k edges
    const int gridPool      = 2048;   // fixed grid, grid-stride over 50k nodes

    // ---------------- Layer 1 ----------------
    k_zero<<<2048, 256, 0, stream>>>(agg, (int)NH);
    for (int r = 0; r < RG_N_REL; ++r) {
        k_gemm_k128<RG_HID><<<gridNodesGemm, gemmBlk, 0, stream>>>(
            h, W1 + (size_t)r * RG_IN_DIM * RG_HID, nullptr, tmp, nullptr,
            RG_N_NODES, 0);
        k_scatter_rel<<<gridScat, 256, 0, stream>>>(tmp, src, dst, rel, agg, r, RG_N_EDGES);
    }
    // self-loop + bias + ReLU fused: h1 = relu(agg + h@Ws1 + b1)
    k_gemm_k128<RG_HID><<<gridNodesGemm, gemmBlk, 0, stream>>>(
        h, Ws1, agg, h1, b1, RG_N_NODES, 1);

    // ---------------- Layer 2 ----------------
    k_zero<<<2048, 256, 0, stream>>>(agg, (int)NH);
    for (int r = 0; r < RG_N_REL; ++r) {
        k_gemm_k128<RG_HID><<<gridNodesGemm, gemmBlk, 0, stream>>>(
            h1, W2 + (size_t)r * RG_HID * RG_HID, nullptr, tmp, nullptr,
            RG_N_NODES, 0);
        k_scatter_rel<<<gridScat, 256, 0, stream>>>(tmp, src, dst, rel, agg, r, RG_N_EDGES);
    }
    // h2 = relu(agg + h1@Ws2 + b2) -> stored in tmp (free after the loop)
    k_gemm_k128<RG_HID><<<gridNodesGemm, gemmBlk, 0, stream>>>(
        h1, Ws2, agg, tmp, b2, RG_N_NODES, 1);

    // ---------------- Mean pooling ----------------
    k_zero<<<64, 256, 0, stream>>>(sums, RG_N_GRAPHS * RG_HID + RG_N_GRAPHS); // sums + cnts contiguous
    k_pool<<<gridPool, 256, 0, stream>>>(tmp, gid, sums, cnts, RG_N_NODES);
    k_pool_div<<<(RG_N_GRAPHS * RG_HID + 255) / 256, 256, 0, stream>>>(sums, cnts, hg);

    // ---------------- FC + ReLU (WMMA) ----------------
    k_gemm_k128<RG_FC><<<gemmGrid(RG_N_GRAPHS, RG_FC), gemmBlk, 0, stream>>>(
        hg, Wfc, nullptr, fcout, bfc, RG_N_GRAPHS, 1);

    // ---------------- Head + softmax ----------------
    k_head<<<RG_N_GRAPHS, 32, 0, stream>>>(fcout, Wc, bc, out);
}